// KNearestNeighbors_6828998001132
// MI455X (gfx1250) — compile-verified
//
#include <hip/hip_runtime.h>

typedef __attribute__((ext_vector_type(2))) float v2f;
typedef __attribute__((ext_vector_type(8))) float v8f;

#define KK 16

__device__ __forceinline__ void topk_insert(float (&best)[KK], int (&bidx)[KK],
                                            float dv, int jv) {
    // best[] kept sorted ascending; guarded insertion with unrolled bubble pass
    if (dv < best[KK - 1]) {
        best[KK - 1] = dv;
        bidx[KK - 1] = jv;
#pragma unroll
        for (int k = KK - 1; k > 0; --k) {
            float da = best[k], db = best[k - 1];
            int ia = bidx[k], ib = bidx[k - 1];
            bool sw = da < db;
            best[k]     = sw ? db : da;
            best[k - 1] = sw ? da : db;
            bidx[k]     = sw ? ib : ia;
            bidx[k - 1] = sw ? ia : ib;
        }
    }
}

// One wave owns 32 query rows (two 16-row WMMA tiles). Loop over a P2 segment in
// 16-column chunks: 2x V_WMMA_F32_16X16X4_F32 -> LDS transpose -> per-lane
// register top-16.  A = [-2x,-2y,-2z,1], B = [bx,by,bz,|b|^2] so D = |b|^2 - 2 a.b;
// |a|^2 added at output (row-constant, order-preserving).
__global__ __launch_bounds__(256) void knn_wmma_kernel(
    const float* __restrict__ p1, const float* __restrict__ p2,
    const int* __restrict__ len1, const int* __restrict__ len2,
    int N, int P1, int P2, int S,
    float* __restrict__ part_d, int* __restrict__ part_i,
    long long* __restrict__ out_idx, float* __restrict__ out_d)
{
    // [wave][col][row(+pad)] : stride 36 floats -> conflict-free, 16B aligned
    __shared__ __align__(16) float tile[8][16][36];

    const int wave = threadIdx.x >> 5;
    const int lane = threadIdx.x & 31;
    const bool lo  = lane < 16;
    const int  c   = lane & 15;
    const int  n   = blockIdx.y;
    const int  seg = blockIdx.z;
    const int  gbase = blockIdx.x * 256 + wave * 32;

    const float* __restrict__ p1n = p1 + (size_t)n * P1 * 3;
    const float* __restrict__ p2n = p2 + (size_t)n * P2 * 3;
    const int L2 = len2[n];
    const int L1 = len1[n];

    // ---- A-matrix registers for the two 16-row tiles ----
    // 16x4 f32 A layout: VGPR0 = (lanes 0-15: K=0 | lanes 16-31: K=2),
    //                    VGPR1 = (K=1 | K=3), M = lane%16.
    v2f a0v, a1v;
    {
        int r0 = gbase + c;       if (r0 >= P1) r0 = P1 - 1;
        float x = p1n[3 * r0 + 0], y = p1n[3 * r0 + 1], z = p1n[3 * r0 + 2];
        a0v[0] = lo ? -2.0f * x : -2.0f * z;
        a0v[1] = lo ? -2.0f * y : 1.0f;
        int r1 = gbase + 16 + c;  if (r1 >= P1) r1 = P1 - 1;
        x = p1n[3 * r1 + 0]; y = p1n[3 * r1 + 1]; z = p1n[3 * r1 + 2];
        a1v[0] = lo ? -2.0f * x : -2.0f * z;
        a1v[1] = lo ? -2.0f * y : 1.0f;
    }

    // ||a||^2 for the row this lane owns (added back at output time)
    const int i = gbase + lane;
    float norm_a;
    {
        int ri = (i < P1) ? i : (P1 - 1);
        float x = p1n[3 * ri], y = p1n[3 * ri + 1], z = p1n[3 * ri + 2];
        norm_a = x * x + y * y + z * z;
    }

    float best[KK];
    int   bidx[KK];
#pragma unroll
    for (int k = 0; k < KK; ++k) { best[k] = __builtin_inff(); bidx[k] = 0; }

    const int segLen = P2 / S;
    const int j0 = seg * segLen, j1 = j0 + segLen;

    for (int jb = j0; jb < j1; jb += 16) {
        // ---- B tile: 4x16 f32. VGPR0 = (K=0 | K=2), VGPR1 = (K=1 | K=3), N = lane%16
        int j = jb + c;
        float bx = p2n[3 * j], by = p2n[3 * j + 1], bz = p2n[3 * j + 2];
        v2f bv;
        bv[0] = lo ? bx : bz;
        bv[1] = lo ? by : (bx * bx + by * by + bz * bz);

        v8f acc = {};
        v8f d0 = __builtin_amdgcn_wmma_f32_16x16x4_f32(
            false, a0v, false, bv, (short)0, acc, false, false);
        v8f d1 = __builtin_amdgcn_wmma_f32_16x16x4_f32(
            false, a1v, false, bv, (short)0, acc, false, false);

        // ---- transpose via LDS: D VGPR r holds row (r | r+8) at col = lane%16.
        // Lane's 8 values are contiguous rows at fixed col -> 2x b128 stores/tile.
        {
            float* t0 = &tile[wave][c][(lo ? 0 : 8)];
            ((float4*)t0)[0] = make_float4(d0[0], d0[1], d0[2], d0[3]);
            ((float4*)t0)[1] = make_float4(d0[4], d0[5], d0[6], d0[7]);
            float* t1 = &tile[wave][c][16 + (lo ? 0 : 8)];
            ((float4*)t1)[0] = make_float4(d1[0], d1[1], d1[2], d1[3]);
            ((float4*)t1)[1] = make_float4(d1[4], d1[5], d1[6], d1[7]);
        }
        // wave-private LDS slice + in-order DS ops within a wave: no barrier needed

        // ---- each lane gathers its own row's 16 candidates (conflict-free)
        float vals[16];
#pragma unroll
        for (int cc = 0; cc < 16; ++cc) vals[cc] = tile[wave][cc][lane];

        const bool allvalid = (jb + 16) <= L2;
#pragma unroll
        for (int cc = 0; cc < 16; ++cc) {
            int jj = jb + cc;
            if (allvalid || (jj < L2)) topk_insert(best, bidx, vals[cc], jj);
        }
    }

    if (i < P1) {
        if (S == 1) {
            size_t base = ((size_t)n * P1 + i) * KK;
            if (i < L1) {
#pragma unroll
                for (int k = 0; k < KK; ++k) {
                    out_idx[base + k] = (long long)bidx[k];
                    out_d[base + k]   = best[k] + norm_a;
                }
            } else {
#pragma unroll
                for (int k = 0; k < KK; ++k) {
                    out_idx[base + k] = 0LL;
                    out_d[base + k]   = 0.0f;
                }
            }
        } else {
            size_t base = (((size_t)seg * N + n) * P1 + i) * KK;
#pragma unroll
            for (int k = 0; k < KK; ++k) {
                part_d[base + k] = best[k] + norm_a;  // inf stays inf for empty segs
                part_i[base + k] = bidx[k];
            }
        }
    }
}

// Merge S per-segment sorted top-16 lists per row; zero padded query rows.
__global__ __launch_bounds__(256) void knn_merge_kernel(
    const float* __restrict__ part_d, const int* __restrict__ part_i,
    const int* __restrict__ len1, int N, int P1, int S,
    long long* __restrict__ out_idx, float* __restrict__ out_d)
{
    size_t gid = (size_t)blockIdx.x * blockDim.x + threadIdx.x;
    if (gid >= (size_t)N * P1) return;
    int n = (int)(gid / P1);
    int i = (int)(gid % P1);
    size_t ob = gid * KK;

    if (i >= len1[n]) {
#pragma unroll
        for (int k = 0; k < KK; ++k) { out_idx[ob + k] = 0LL; out_d[ob + k] = 0.0f; }
        return;
    }

    float best[KK];
    int   bidx[KK];
#pragma unroll
    for (int k = 0; k < KK; ++k) { best[k] = __builtin_inff(); bidx[k] = 0; }

    for (int s = 0; s < S; ++s) {
        size_t pb = (((size_t)s * N + n) * P1 + i) * KK;
#pragma unroll
        for (int k = 0; k < KK; ++k) {
            float dv = part_d[pb + k];
            int   jv = part_i[pb + k];
            topk_insert(best, bidx, dv, jv);
        }
    }
#pragma unroll
    for (int k = 0; k < KK; ++k) {
        out_idx[ob + k] = (long long)bidx[k];
        out_d[ob + k]   = best[k];
    }
}

extern "C" void kernel_launch(void* const* d_in, const int* in_sizes, int n_in,
                              void* d_out, int out_size, void* d_ws, size_t ws_size,
                              hipStream_t stream) {
    const float* p1   = (const float*)d_in[0];
    const float* p2   = (const float*)d_in[1];
    const int*   len1 = (const int*)d_in[2];
    const int*   len2 = (const int*)d_in[3];

    const int N  = in_sizes[2];
    const int P1 = in_sizes[0] / (N * 3);
    const int P2 = in_sizes[1] / (N * 3);
    const int K  = KK;

    const size_t rows = (size_t)N * P1;
    long long* out_idx = (long long*)d_out;                 // (N,P1,K) int64 first
    float*     out_d   = (float*)(out_idx + rows * K);      // then (N,P1,K) f32

    // P2-split factor: more waves in flight, merged afterwards; bounded by ws_size
    const size_t segBytes = rows * K * 8;  // f32 dist + i32 idx partials / segment
    int S = 1;
    if (ws_size >= 4 * segBytes && (P2 % 64) == 0) S = 4;
    else if (ws_size >= 2 * segBytes && (P2 % 32) == 0) S = 2;

    float* part_d = (float*)d_ws;
    int*   part_i = (int*)((char*)d_ws + (size_t)S * rows * K * 4);

    dim3 grid((P1 + 255) / 256, N, S);
    knn_wmma_kernel<<<grid, 256, 0, stream>>>(p1, p2, len1, len2,
                                              N, P1, P2, S,
                                              part_d, part_i, out_idx, out_d);
    if (S > 1) {
        int threads = 256;
        int blocks  = (int)((rows + threads - 1) / threads);
        knn_merge_kernel<<<blocks, threads, 0, stream>>>(part_d, part_i, len1,
                                                         N, P1, S, out_idx, out_d);
    }
}